// MentionScorerGap_2482491097282
// MI455X (gfx1250) — compile-verified
//
#include <hip/hip_runtime.h>
#include <hip/hip_bf16.h>

typedef float v2f __attribute__((ext_vector_type(2)));
typedef float v8f __attribute__((ext_vector_type(8)));

#define T_TOK 4096
#define D_EMB 768
#define H_HID 150
#define L_SPAN 10
#define N_SPAN (T_TOK * L_SPAN)
#define PCOLS 160   // padded column stride for the P tables (10 x 16-col tiles)

// ---------------------------------------------------------------------------
// Kernel 1: fused fp32 WMMA GEMM
//   C = E[4096x768] @ [ W_a1 | W_m1[0:768,:] | W_m1[768:1536,:] | W_m1[1536:2304,:] ]
// Each weight block is [768 x 150]; outputs go to 4 tables of [4096 x 160]
// (cols 150..159 forced to zero via a B-operand mask so no OOB weight reads).
// One wave computes one 16x16 output tile; K-loop = 192 x v_wmma_f32_16x16x4_f32.
// ---------------------------------------------------------------------------
__global__ __launch_bounds__(256)
void gemm_wmma_f32(const float* __restrict__ E,
                   const float* __restrict__ Wa1,
                   const float* __restrict__ Wm1,
                   float* __restrict__ P /* 4 x 4096 x 160 */) {
  const int lane = threadIdx.x & 31;
  const int wid  = (blockIdx.x << 3) | (threadIdx.x >> 5);  // global wave id
  const int tn = wid % 40;          // 40 column tiles (4 groups x 160/16)
  const int tm = wid / 40;          // 256 row tiles
  const int ncolG = tn * 16;
  const int group = ncolG / PCOLS;        // which weight block (0..3)
  const int nloc  = ncolG % PCOLS;        // tile base column within block
  const float* Wg = (group == 0) ? Wa1
                                 : (Wm1 + (size_t)(group - 1) * D_EMB * H_HID);
  float* Pg = P + (size_t)group * T_TOK * PCOLS;

  // A fragment layout (16x4 f32): lanes 0-15 = rows M, VGPR0=K+0 / VGPR1=K+1;
  // lanes 16-31 = rows M, VGPR0=K+2 / VGPR1=K+3.
  const int row   = lane & 15;
  const int khalf = (lane >> 4) << 1;      // 0 or 2
  // B fragment (4x16 f32), mirrored: lane carries column nloc+(lane&15).
  const int col   = nloc + (lane & 15);
  const float colmask = (col < H_HID) ? 1.0f : 0.0f;  // zero padded cols
  const int   colc    = (col < H_HID) ? col : (H_HID - 1);

  const float* Arow = E + (size_t)(tm * 16 + row) * D_EMB + khalf;
  const float* Bcol = Wg + colc;

  v8f c = {0.f, 0.f, 0.f, 0.f, 0.f, 0.f, 0.f, 0.f};
#pragma unroll 4
  for (int k = 0; k < D_EMB; k += 4) {
    v2f a;
    a.x = Arow[k];
    a.y = Arow[k + 1];
    v2f b;
    b.x = Bcol[(size_t)(k + khalf) * H_HID] * colmask;
    b.y = Bcol[(size_t)(k + khalf + 1) * H_HID] * colmask;
    c = __builtin_amdgcn_wmma_f32_16x16x4_f32(false, a, false, b,
                                              (short)0, c, false, false);
  }

  // C layout: VGPR j -> M = j (lanes 0-15) / M = 8+j (lanes 16-31), N = lane&15
  const int hi = lane >> 4;
#pragma unroll
  for (int j = 0; j < 8; ++j) {
    const int m = tm * 16 + j + (hi << 3);
    Pg[(size_t)m * PCOLS + (nloc + (lane & 15))] = c[j];
  }
}

// ---------------------------------------------------------------------------
// Kernel 2: per-token attention logits
//   attns[t] = relu(Pa[t,:] + b_a1) . W_a2 + b_a2
// ---------------------------------------------------------------------------
__global__ __launch_bounds__(256)
void attn_logits(const float* __restrict__ Pa, const float* __restrict__ b_a1,
                 const float* __restrict__ W_a2, const float* __restrict__ b_a2,
                 float* __restrict__ attns) {
  const int t = blockIdx.x * blockDim.x + threadIdx.x;
  if (t >= T_TOK) return;
  const float* rowp = Pa + (size_t)t * PCOLS;
  float acc = b_a2[0];
#pragma unroll 5
  for (int h = 0; h < H_HID; ++h) {
    float x = rowp[h] + b_a1[h];
    acc = fmaf(fmaxf(x, 0.0f), W_a2[h], acc);
  }
  attns[t] = acc;
}

// ---------------------------------------------------------------------------
// Kernel 3: one block (192 threads, 6 waves) per span.
//   - per-thread 10-way softmax of attention logits (redundant but barrier-free)
//   - float4 coalesced writes of g_i = [E[s], E[e], sum_l w_l E[idx_l]]
//   - mention score from factored tables: relu(P1[s]+P2[e]+sum w P3[idx]+b).W_m2
// ---------------------------------------------------------------------------
__global__ __launch_bounds__(192)
void span_kernel(const float* __restrict__ E,
                 const int* __restrict__ starts,
                 const int* __restrict__ ends,
                 const float* __restrict__ attns,
                 const float* __restrict__ P1,
                 const float* __restrict__ P2,
                 const float* __restrict__ P3,
                 const float* __restrict__ b_m1,
                 const float* __restrict__ W_m2,
                 const float* __restrict__ b_m2,
                 float* __restrict__ out_g,   // [N, 2304]
                 float* __restrict__ out_s) { // [N]
  const int n = blockIdx.x;
  const int tid = threadIdx.x;
  const int s = starts[n];
  const int e = ends[n];

  int   idx[L_SPAN];
  float w[L_SPAN];
  float mx = -3.0e38f;
#pragma unroll
  for (int l = 0; l < L_SPAN; ++l) {
    const int p = s + l;
    idx[l] = (p < T_TOK) ? p : (T_TOK - 1);
    const float a = (p <= e) ? attns[idx[l]] : -1.0e10f;
    w[l] = a;
    mx = fmaxf(mx, a);
  }
  float sum = 0.f;
#pragma unroll
  for (int l = 0; l < L_SPAN; ++l) { w[l] = __expf(w[l] - mx); sum += w[l]; }
  const float inv = 1.0f / sum;
#pragma unroll
  for (int l = 0; l < L_SPAN; ++l) w[l] *= inv;

  // g_i row: 2304 floats = 3 sections x 192 float4 -> one b128 per thread each
  const float4* E4 = (const float4*)E;
  float4* G4 = (float4*)(out_g + (size_t)n * (3 * D_EMB));
  const int R4 = D_EMB / 4;  // 192
  const float4 vs = E4[(size_t)s * R4 + tid];
  const float4 ve = E4[(size_t)e * R4 + tid];
  float4 acc = make_float4(0.f, 0.f, 0.f, 0.f);
#pragma unroll
  for (int l = 0; l < L_SPAN; ++l) {
    const float4 v = E4[(size_t)idx[l] * R4 + tid];
    acc.x = fmaf(w[l], v.x, acc.x);
    acc.y = fmaf(w[l], v.y, acc.y);
    acc.z = fmaf(w[l], v.z, acc.z);
    acc.w = fmaf(w[l], v.w, acc.w);
  }
  G4[tid]           = vs;
  G4[R4 + tid]      = ve;
  G4[2 * R4 + tid]  = acc;

  // mention score via factored hidden layer
  __shared__ float red[192];
  float part = 0.f;
  if (tid < H_HID) {
    float hid = P1[(size_t)s * PCOLS + tid] + P2[(size_t)e * PCOLS + tid] +
                b_m1[tid];
#pragma unroll
    for (int l = 0; l < L_SPAN; ++l)
      hid = fmaf(w[l], P3[(size_t)idx[l] * PCOLS + tid], hid);
    part = fmaxf(hid, 0.0f) * W_m2[tid];
  }
  red[tid] = part;
  __syncthreads();
  if (tid < 96) red[tid] += red[tid + 96];
  __syncthreads();
  if (tid < 32) {
    float v = red[tid] + red[tid + 32] + red[tid + 64];
#pragma unroll
    for (int off = 16; off > 0; off >>= 1) v += __shfl_down(v, off);
    if (tid == 0) out_s[n] = v + b_m2[0];
  }
}

// ---------------------------------------------------------------------------
extern "C" void kernel_launch(void* const* d_in, const int* in_sizes, int n_in,
                              void* d_out, int out_size, void* d_ws,
                              size_t ws_size, hipStream_t stream) {
  const float* embeds = (const float*)d_in[0];
  const int*   starts = (const int*)d_in[1];
  const int*   ends   = (const int*)d_in[2];
  const float* W_a1   = (const float*)d_in[3];
  const float* b_a1   = (const float*)d_in[4];
  const float* W_a2   = (const float*)d_in[5];
  const float* b_a2   = (const float*)d_in[6];
  const float* W_m1   = (const float*)d_in[7];
  const float* b_m1   = (const float*)d_in[8];
  const float* W_m2   = (const float*)d_in[9];
  const float* b_m2   = (const float*)d_in[10];

  float* out = (float*)d_out;
  float* ws  = (float*)d_ws;

  float* P     = ws;                              // 4 * 4096 * 160 floats
  float* attns = ws + (size_t)4 * T_TOK * PCOLS;  // 4096 floats

  // 256 row-tiles x 40 col-tiles = 10240 waves, 8 waves per block
  gemm_wmma_f32<<<1280, 256, 0, stream>>>(embeds, W_a1, W_m1, P);

  attn_logits<<<(T_TOK + 255) / 256, 256, 0, stream>>>(P, b_a1, W_a2, b_a2,
                                                       attns);

  span_kernel<<<N_SPAN, 192, 0, stream>>>(
      embeds, starts, ends, attns,
      P + (size_t)1 * T_TOK * PCOLS,   // P1: embeds @ W_m1[0:768]
      P + (size_t)2 * T_TOK * PCOLS,   // P2: embeds @ W_m1[768:1536]
      P + (size_t)3 * T_TOK * PCOLS,   // P3: embeds @ W_m1[1536:2304]
      b_m1, W_m2, b_m2,
      out, out + (size_t)N_SPAN * 3 * D_EMB);
}